// KPMiniMod_47957604827526
// MI455X (gfx1250) — compile-verified
//
#include <hip/hip_runtime.h>

// ---------------------------------------------------------------------------
// KPMiniMod for MI455X (gfx1250, wave32)
//   phase A: mod = sigmoid(leaky(s_feats@w1+b1)@w2)  via V_WMMA_F32_16X16X4_F32
//   phase B: geometry argmin + modulated gather-reduce (L2-resident gather)
// ---------------------------------------------------------------------------

typedef __attribute__((ext_vector_type(2))) float v2f;
typedef __attribute__((ext_vector_type(8))) float v8f;

#define MQ    32000
#define HNB   32
#define CCH   64
#define KPT   15
#define CO    120   // K * CH_PER_GRP

__device__ __forceinline__ float leaky01(float x) { return x >= 0.0f ? x : 0.1f * x; }
__device__ __forceinline__ float sigmoidf(float x) { return 1.0f / (1.0f + __expf(-x)); }

// ---------------------------------------------------------------------------
// Phase A: fused GEMM1 (64x64) -> leaky -> GEMM2 (64x120) -> sigmoid -> mod
// One wave per 16-row strip of M.  f32 WMMA 16x16x4, K stepped by 4.
//   A 16x4 layout : lane l -> row = l&15 ; v0/v1 = K = 2*(l>>4) + {0,1}
//   B 4x16 layout : lane l -> col = l&15 ; v0/v1 = K = 2*(l>>4) + {0,1}
//   C/D 16x16     : lane l, elem j -> row = j + 8*(l>>4), col = l&15
// ---------------------------------------------------------------------------
__global__ __launch_bounds__(256) void kpm_mod_gemm(
    const float* __restrict__ X,    // s_feats (M,64)
    const float* __restrict__ W1,   // (64,64)
    const float* __restrict__ B1v,  // (64,)
    const float* __restrict__ W2,   // (64,120)
    float* __restrict__ modout)     // (M,120)
{
    __shared__ float hid[8][16 * 65];   // per-wave 16x64 hidden tile, padded

    const int wave = threadIdx.x >> 5;
    const int lane = threadIdx.x & 31;
    const int lo   = lane & 15;
    const int hi   = lane >> 4;
    const int m0   = (blockIdx.x * 8 + wave) * 16;

    // ---- GEMM1: hidden = leaky(X @ W1 + b1) -------------------------------
    v8f acc[4];
#pragma unroll
    for (int nt = 0; nt < 4; ++nt)
#pragma unroll
        for (int j = 0; j < 8; ++j) acc[nt][j] = 0.0f;

#pragma unroll
    for (int kb = 0; kb < 16; ++kb) {
        const int k0 = kb * 4 + 2 * hi;
        const float* ap = X + (size_t)(m0 + lo) * CCH + k0;
        v2f a; a.x = ap[0]; a.y = ap[1];
#pragma unroll
        for (int nt = 0; nt < 4; ++nt) {
            const int col = nt * 16 + lo;
            v2f b;
            b.x = W1[(size_t)k0 * CCH + col];
            b.y = W1[(size_t)(k0 + 1) * CCH + col];
            acc[nt] = __builtin_amdgcn_wmma_f32_16x16x4_f32(
                false, a, false, b, (short)0, acc[nt], false, false);
        }
    }

    // bias + leaky -> LDS (row-major so we can re-read in A-frag layout)
    float* hb = &hid[wave][0];
#pragma unroll
    for (int nt = 0; nt < 4; ++nt) {
        const int   col  = nt * 16 + lo;
        const float bias = B1v[col];
#pragma unroll
        for (int j = 0; j < 8; ++j) {
            const int row = j + 8 * hi;
            hb[row * 65 + col] = leaky01(acc[nt][j] + bias);
        }
    }
    __syncthreads();

    // ---- GEMM2: mod = sigmoid(hidden @ W2), N = 120 padded to 128 ---------
    v8f acc2[8];
#pragma unroll
    for (int nt = 0; nt < 8; ++nt)
#pragma unroll
        for (int j = 0; j < 8; ++j) acc2[nt][j] = 0.0f;

#pragma unroll
    for (int kb = 0; kb < 16; ++kb) {
        const int k0 = kb * 4 + 2 * hi;
        v2f a;
        a.x = hb[lo * 65 + k0];
        a.y = hb[lo * 65 + k0 + 1];
#pragma unroll
        for (int nt = 0; nt < 8; ++nt) {
            const int col = nt * 16 + lo;
            v2f b;
            if (col < CO) {
                b.x = W2[(size_t)k0 * CO + col];
                b.y = W2[(size_t)(k0 + 1) * CO + col];
            } else {
                b.x = 0.0f; b.y = 0.0f;
            }
            acc2[nt] = __builtin_amdgcn_wmma_f32_16x16x4_f32(
                false, a, false, b, (short)0, acc2[nt], false, false);
        }
    }

#pragma unroll
    for (int nt = 0; nt < 8; ++nt) {
        const int col = nt * 16 + lo;
        if (col < CO) {
#pragma unroll
            for (int j = 0; j < 8; ++j) {
                const int row = j + 8 * hi;
                modout[(size_t)(m0 + row) * CO + col] = sigmoidf(acc2[nt][j]);
            }
        }
    }
}

// ---------------------------------------------------------------------------
// Phase B: per-query wave.  Lane h owns neighbor h (argmin over 15 kernel
// points), then uniform h-loop broadcasts (idx, k, infl) with wave32 shuffles
// and each lane accumulates 2 channels from the L2-resident feature table.
// ---------------------------------------------------------------------------
__global__ __launch_bounds__(256) void kpm_gather_agg(
    const float* __restrict__ qp,    // (M,3)
    const float* __restrict__ sp,    // (N,3)
    const float* __restrict__ sf,    // (N,64)
    const int*   __restrict__ nidx,  // (M,32)
    const float* __restrict__ kp,    // (15,3)
    const float* __restrict__ wts,   // (15,64)
    const float* __restrict__ modv,  // (M,120)
    float* __restrict__ out)         // (M,64)
{
    __shared__ float s_kp[KPT * 3];
    __shared__ float s_w[KPT * CCH];

    const int tid = threadIdx.x;
    for (int i = tid; i < KPT * 3; i += 256)   s_kp[i] = kp[i];
    for (int i = tid; i < KPT * CCH; i += 256) s_w[i]  = wts[i];
    __syncthreads();

    const int wave = tid >> 5;
    const int lane = tid & 31;
    const int m    = blockIdx.x * 8 + wave;

    const float qx = qp[(size_t)m * 3 + 0];
    const float qy = qp[(size_t)m * 3 + 1];
    const float qz = qp[(size_t)m * 3 + 2];

    // lane = neighbor index h
    const int   si = nidx[(size_t)m * HNB + lane];
    const float px = sp[(size_t)si * 3 + 0] - qx;
    const float py = sp[(size_t)si * 3 + 1] - qy;
    const float pz = sp[(size_t)si * 3 + 2] - qz;

    float best = 1.0e30f;
    int   bk   = 0;
#pragma unroll
    for (int k = 0; k < KPT; ++k) {
        const float dx = px - s_kp[k * 3 + 0];
        const float dy = py - s_kp[k * 3 + 1];
        const float dz = pz - s_kp[k * 3 + 2];
        const float d2 = dx * dx + dy * dy + dz * dz;
        if (d2 < best) { best = d2; bk = k; }   // strict <  == first argmin
    }
    float infl = 1.0f - sqrtf(best);            // SIGMA = 1
    infl = fmaxf(infl, 0.0f);

    const int c0 = lane * 2;        // two channels per lane
    const int cg = lane >> 2;       // channel-group (c/8), same for both chans
    float ax = 0.0f, ay = 0.0f;

    for (int h = 0; h < HNB; ++h) {
        const float wgt = __shfl(infl, h, 32);  // wave-uniform
        if (wgt > 0.0f) {
            const int nh = __shfl(si, h, 32);
            const int kh = __shfl(bk, h, 32);
            const float2 f = *(const float2*)(sf + (size_t)nh * CCH + c0);
            const float wx = s_w[kh * CCH + c0];
            const float wy = s_w[kh * CCH + c0 + 1];
            const float mv = modv[(size_t)m * CO + kh * 8 + cg];
            const float s  = mv * wgt;
            ax = fmaf(f.x * wx, s, ax);
            ay = fmaf(f.y * wy, s, ay);
        }
    }

    float2 o; o.x = ax; o.y = ay;
    *(float2*)(out + (size_t)m * CCH + c0) = o;
}

// ---------------------------------------------------------------------------
extern "C" void kernel_launch(void* const* d_in, const int* in_sizes, int n_in,
                              void* d_out, int out_size, void* d_ws, size_t ws_size,
                              hipStream_t stream) {
    (void)in_sizes; (void)n_in; (void)out_size; (void)ws_size;
    const float* qp  = (const float*)d_in[0];  // q_pts        (M,3)
    const float* sp  = (const float*)d_in[1];  // s_pts        (N,3)
    const float* sf  = (const float*)d_in[2];  // s_feats      (N,64)
    const int*   nid = (const int*)  d_in[3];  // neighb_inds  (M,32)
    const float* kp  = (const float*)d_in[4];  // kernel_points(15,3)
    const float* wts = (const float*)d_in[5];  // weights      (15,64)
    const float* w1  = (const float*)d_in[6];  // (64,64)
    const float* b1  = (const float*)d_in[7];  // (64,)
    const float* w2  = (const float*)d_in[8];  // (64,120)
    float* out    = (float*)d_out;
    float* modbuf = (float*)d_ws;              // (M,120) = 15.36 MB scratch

    // Phase A: 2000 strips of 16 rows, 8 waves/block -> 250 blocks
    kpm_mod_gemm<<<250, 256, 0, stream>>>(sf, w1, b1, w2, modbuf);
    // Phase B: 1 wave per query, 8 queries/block -> 4000 blocks
    kpm_gather_agg<<<4000, 256, 0, stream>>>(qp, sp, sf, nid, kp, wts, modbuf, out);
}